// RoIPoolingLayer_56822417326275
// MI455X (gfx1250) — compile-verified
//
#include <hip/hip_runtime.h>
#include <stdint.h>

// RoI max-pooling, MI455X (gfx1250).
// img : (1, 512, 512, 256) fp32 NHWC
// rois: (1, 256, 4) int32 (x, y, w=28, h=28)
// out : (1, 256, 7, 7, 256) fp32
//
// Bandwidth-bound (~218 MB total -> ~9.4 us at 23.3 TB/s). Strategy:
//  - 1 block per (roi, pooled-row p). Each block streams 4 contiguous
//    28*256-float image-row slices through a double-buffered LDS staging
//    area using gfx1250 async global->LDS DMA (ASYNCcnt), overlapping the
//    DMA of row r+1 with the max-reduction of row r.
//  - thread t owns channel t: conflict-free ds_load_b32, v_max_num_f32.

#define POOLK   7
#define ROIW    28
#define IMG_W   512
#define IMG_C   256
#define NROI    256

#define ROW_FLOATS (ROIW * IMG_C)   // 7168 floats = 28672 bytes per row slice
#define ROW_F4     (ROW_FLOATS / 4) // 1792 float4 transfers per row slice

// Issue one row-slice (28*256 floats, contiguous in memory) as async
// global->LDS B128 copies. GVS mode: SADDR = uniform 64-bit row base,
// VADDR = per-lane 32-bit byte offset, VDST = per-lane LDS byte address.
__device__ __forceinline__ void async_copy_row(const float* rowPtr,
                                               unsigned ldsBase, int t) {
#pragma unroll
  for (int i = 0; i < ROW_F4 / 256; ++i) {  // 7 x B128 per thread
    unsigned byteOff = (unsigned)(t + i * 256) * 16u;
    unsigned ldsAddr = ldsBase + byteOff;
    asm volatile("global_load_async_to_lds_b128 %0, %1, %2"
                 :
                 : "v"(ldsAddr), "v"(byteOff), "s"(rowPtr)
                 : "memory");
  }
}

__device__ __forceinline__ void wait_async_all() {
  asm volatile("s_wait_asynccnt 0" ::: "memory");
}

__global__ __launch_bounds__(256) void roi_pool_async_kernel(
    const float* __restrict__ img, const int* __restrict__ rois,
    float* __restrict__ out) {
  __shared__ float buf[2][ROW_FLOATS];  // 2 x 28 KB double buffer

  const int t = threadIdx.x;   // channel 0..255
  const int p = blockIdx.x;    // pooled row 0..6
  const int r = blockIdx.y;    // roi 0..255

  // RoI origin; force uniform so the DMA base lives in SGPRs.
  const int x0 = __builtin_amdgcn_readfirstlane(rois[r * 4 + 0]);
  const int y0 = __builtin_amdgcn_readfirstlane(rois[r * 4 + 1]);

  // LDS byte offsets of the two staging buffers (flat LDS addr low 32 bits
  // == wave-relative LDS byte address per the gfx1250 aperture scheme).
  const unsigned ldsBase0 = (unsigned)(uintptr_t)(void*)&buf[0][0];
  const unsigned ldsBase1 = (unsigned)(uintptr_t)(void*)&buf[1][0];

  const size_t rowStride = (size_t)IMG_W * IMG_C;  // floats per image row
  const float* row0 = img + ((size_t)(x0 + p * 4) * IMG_W + y0) * IMG_C;

  float acc[POOLK];
#pragma unroll
  for (int q = 0; q < POOLK; ++q) acc[q] = -__builtin_inff();

  // Prologue: stage row 0.
  async_copy_row(row0, ldsBase0, t);
  wait_async_all();
  __syncthreads();

#pragma unroll
  for (int rr = 0; rr < 4; ++rr) {
    // Kick off DMA of the next row into the other buffer. Safe: that
    // buffer was last read before the barrier at the end of iter rr-1.
    if (rr + 1 < 4)
      async_copy_row(row0 + (size_t)(rr + 1) * rowStride,
                     (rr & 1) ? ldsBase0 : ldsBase1, t);

    // Max-reduce current row from LDS: thread t reads 28 cols of its
    // channel; lane->consecutive banks, conflict-free.
    const float* cur = &buf[rr & 1][0];
#pragma unroll
    for (int j = 0; j < ROIW; ++j)
      acc[j >> 2] = fmaxf(acc[j >> 2], cur[j * IMG_C + t]);

    if (rr + 1 < 4) {
      wait_async_all();   // next row fully landed (per-wave)
      __syncthreads();    // ... and for every wave in the block
    }
  }

  // out[((r*7 + p)*7 + q)*256 + t] — contiguous 256-ch stores per bin.
  float* o = out + (((size_t)r * POOLK + p) * POOLK) * IMG_C + t;
#pragma unroll
  for (int q = 0; q < POOLK; ++q) o[q * IMG_C] = acc[q];
}

extern "C" void kernel_launch(void* const* d_in, const int* in_sizes, int n_in,
                              void* d_out, int out_size, void* d_ws,
                              size_t ws_size, hipStream_t stream) {
  const float* img = (const float*)d_in[0];
  const int* rois = (const int*)d_in[1];
  float* out = (float*)d_out;

  dim3 grid(POOLK, NROI);  // 7 pooled rows x 256 rois
  roi_pool_async_kernel<<<grid, 256, 0, stream>>>(img, rois, out);
}